// NeighborAggregation_37417755082987
// MI455X (gfx1250) — compile-verified
//
#include <hip/hip_runtime.h>
#include <hip/hip_bf16.h>

// ---------------------------------------------------------------------------
// Fused GNN neighbor-aggregation layer for MI455X (gfx1250, wave32).
//   agg   = mean(valid neighbor rows)           -- gather, L2-bound
//   trans = agg @ W^T + b                       -- fp32 WMMA 16x16x4
//   out   = relu(LN(x + trans)) or passthrough  -- wave32 shuffle LN
// Block = 128 threads (4 waves); each wave owns a 16-node tile (64 nodes/blk).
// ---------------------------------------------------------------------------

typedef __attribute__((ext_vector_type(2))) float v2f;
typedef __attribute__((ext_vector_type(8))) float v8f;

#define D128   128
#define PADS   132                 // padded LDS row stride (floats): %64==4 -> conflict-free, 16B aligned
#define NWAVES 4
#define TILE   16

__global__ void NeighborAggregation_37417755082987_kernel(
    const float* __restrict__ nf,      // [N,128]
    const int*   __restrict__ nidx,    // [N,K]
    const int*   __restrict__ ncnt,    // [N]
    const float* __restrict__ W,       // [128,128]
    const float* __restrict__ bias,    // [128]
    const float* __restrict__ lnw,     // [128]
    const float* __restrict__ lnb,     // [128]
    float*       __restrict__ out,     // [N,128]
    int N, int Kn)
{
    extern __shared__ float lds[];
    float* wt   = lds;                                   // Wt[128][PADS]  (Wt[k][c] = W[c][k])
    float* aggb = lds + D128 * PADS;                     // [NWAVES][16][PADS]
    float* trb  = aggb + NWAVES * TILE * PADS;           // [NWAVES][16][PADS]

    const int tid  = threadIdx.x;
    const int w    = tid >> 5;
    const int lane = tid & 31;
    const int m    = lane & 15;        // row-within-halfwave
    const int h    = lane >> 4;        // halfwave id

    // ---- Stage W transposed into LDS (coalesced global reads) -------------
    for (int r = 0; r < D128; ++r)
        wt[tid * PADS + r] = W[r * D128 + tid];          // Wt[tid][r] = W[r][tid]
    __syncthreads();

    const float4* nf4 = reinterpret_cast<const float4*>(nf);
    float* agg = aggb + w * TILE * PADS;
    float* tr  = trb  + w * TILE * PADS;
    const int tileBase = blockIdx.x * (NWAVES * TILE) + w * TILE;

    // ---- Phase 1: mean aggregation into LDS A-tile (16 x 128 f32) ---------
    for (int i = 0; i < TILE; ++i) {
        int node = tileBase + i;
        if (node >= N) node = N - 1;                     // clamp; stores masked later
        const int cnt = ncnt[node];
        const int* ip = nidx + (long long)node * Kn;
        float4 acc = make_float4(0.f, 0.f, 0.f, 0.f);
        for (int k = 0; k < cnt; ++k) {
            const int j = ip[k];
            if (k + 1 < cnt)                              // gfx1250 global_prefetch
                __builtin_prefetch(&nf4[(long long)ip[k + 1] * 32 + lane], 0, 0);
            const float4 v = nf4[(long long)j * 32 + lane];
            acc.x += v.x; acc.y += v.y; acc.z += v.z; acc.w += v.w;
        }
        const float inv = 1.0f / (float)(cnt > 0 ? cnt : 1);
        float* arow = agg + i * PADS + lane * 4;
        arow[0] = acc.x * inv; arow[1] = acc.y * inv;
        arow[2] = acc.z * inv; arow[3] = acc.w * inv;
    }
    __syncthreads();

    // ---- Phase 2: transformed = agg @ W^T + b via V_WMMA_F32_16X16X4_F32 --
    // A frag (16x4 f32): lanes 0-15 hold M=lane {K0,K1}; lanes 16-31 {K2,K3}.
    // B frag (4x16 f32): mirrored -> rows of Wt, 16 contiguous floats/halfwave.
    // C/D (16x16 f32): VGPR v -> row v + 8*h, lane&15 -> col.
    for (int c = 0; c < 8; ++c) {
        const int col = c * 16 + m;
        v8f acc = {0.f, 0.f, 0.f, 0.f, 0.f, 0.f, 0.f, 0.f};
#pragma unroll 8
        for (int kt = 0; kt < 32; ++kt) {
            const int kk = kt * 4 + 2 * h;
            v2f a, bf;
            a.x  = agg[m * PADS + kk];
            a.y  = agg[m * PADS + kk + 1];
            bf.x = wt[kk * PADS + col];
            bf.y = wt[(kk + 1) * PADS + col];
            acc = __builtin_amdgcn_wmma_f32_16x16x4_f32(
                /*neg_a=*/false, a, /*neg_b=*/false, bf,
                /*c_mod=*/(short)0, acc, /*reuse_a=*/false, /*reuse_b=*/false);
        }
        const float bv = bias[col];
#pragma unroll
        for (int v = 0; v < 8; ++v) {
            const int row = v + 8 * h;
            tr[row * PADS + col] = acc[v] + bv;
        }
    }
    __syncthreads();

    // ---- Phase 3: residual + LayerNorm (wave32 shuffle) + ReLU + select ---
    const float4 g4 = reinterpret_cast<const float4*>(lnw)[lane];
    const float4 e4 = reinterpret_cast<const float4*>(lnb)[lane];
    for (int i = 0; i < TILE; ++i) {
        const int node = tileBase + i;
        if (node >= N) break;                            // wave-uniform exit
        const float4 x = nf4[(long long)node * 32 + lane];
        const float* trow = tr + i * PADS + lane * 4;
        const float4 t = *reinterpret_cast<const float4*>(trow);
        float4 cb = make_float4(x.x + t.x, x.y + t.y, x.z + t.z, x.w + t.w);

        float s = cb.x + cb.y + cb.z + cb.w;
        for (int off = 16; off > 0; off >>= 1) s += __shfl_xor(s, off, 32);
        const float mu = s * (1.0f / 128.0f);

        const float dx = cb.x - mu, dy = cb.y - mu, dz = cb.z - mu, dw = cb.w - mu;
        float ss = dx * dx + dy * dy + dz * dz + dw * dw;
        for (int off = 16; off > 0; off >>= 1) ss += __shfl_xor(ss, off, 32);
        const float rstd = rsqrtf(ss * (1.0f / 128.0f) + 1e-5f);

        float4 y;
        y.x = fmaxf(dx * rstd * g4.x + e4.x, 0.f);
        y.y = fmaxf(dy * rstd * g4.y + e4.y, 0.f);
        y.z = fmaxf(dz * rstd * g4.z + e4.z, 0.f);
        y.w = fmaxf(dw * rstd * g4.w + e4.w, 0.f);

        const float4 o = (ncnt[node] > 0) ? y : x;
        reinterpret_cast<float4*>(out)[(long long)node * 32 + lane] = o;
    }
}

extern "C" void kernel_launch(void* const* d_in, const int* in_sizes, int n_in,
                              void* d_out, int out_size, void* d_ws, size_t ws_size,
                              hipStream_t stream) {
    const float* nf   = (const float*)d_in[0];
    const int*   nidx = (const int*)  d_in[1];
    const int*   ncnt = (const int*)  d_in[2];
    const float* W    = (const float*)d_in[3];
    const float* bias = (const float*)d_in[4];
    const float* lnw  = (const float*)d_in[5];
    const float* lnb  = (const float*)d_in[6];
    float* out = (float*)d_out;

    const int N  = in_sizes[0] / D128;
    const int Kn = in_sizes[1] / N;

    const int nodesPerBlock = NWAVES * TILE;             // 64
    const int grid = (N + nodesPerBlock - 1) / nodesPerBlock;
    const size_t shmem = (size_t)(D128 * PADS + 2 * NWAVES * TILE * PADS) * sizeof(float);

    NeighborAggregation_37417755082987_kernel<<<grid, NWAVES * 32, shmem, stream>>>(
        nf, nidx, ncnt, W, bias, lnw, lnb, out, N, Kn);
}